// Graph_encoder_42666205119185
// MI455X (gfx1250) — compile-verified
//
#include <hip/hip_runtime.h>
#include <hip/hip_bf16.h>

typedef __attribute__((ext_vector_type(2))) float v2f;
typedef __attribute__((ext_vector_type(8))) float v8f;

// ---------------------------------------------------------------------------
// Tiled fp32 GEMM using V_WMMA_F32_16X16X4_F32.
//   C[M,N] = A[M,K] @ B[K,N] + bias[N]
// Block: 128 threads (4 waves). Block tile 64(M) x 64(N). K staged in LDS,
// 16 at a time. Wave w computes rows [bm+16w, bm+16w+16) x all 64 cols
// (four 16x16 WMMA accumulators).
// LDS strides: As row stride 20 floats (16B aligned float4 stores, and the
// stride-20 fragment reads hit 16 distinct banks per half-wave, with the two
// halves in different residue classes mod 4 -> conflict-free).
// Bs row stride 72 floats (16B aligned; halves read rows kk and kk+2 whose
// bank windows are offset by 16 -> disjoint).
// ---------------------------------------------------------------------------
__global__ __launch_bounds__(128)
void gemm_bias_wmma_f32(const float* __restrict__ A, const float* __restrict__ B,
                        const float* __restrict__ bias, float* __restrict__ C,
                        int M, int N, int K)
{
    __shared__ float As[64 * 20];
    __shared__ float Bs[16 * 72];

    const int t    = threadIdx.x;
    const int wave = t >> 5;
    const int lane = t & 31;
    const int half = lane >> 4;   // 0: lanes 0-15, 1: lanes 16-31
    const int lm   = lane & 15;

    const int bm = blockIdx.x * 64;
    const int bn = blockIdx.y * 64;

    // Accumulators: 4 subtiles of 16x16, initialized with broadcast bias.
    v8f acc[4];
#pragma unroll
    for (int s = 0; s < 4; ++s) {
        const float bv = bias[bn + s * 16 + lm];
#pragma unroll
        for (int r = 0; r < 8; ++r) acc[s][r] = bv;
    }

    const int aRowBase = wave * 16 + lm;   // LDS row this lane reads for A frags

    for (int k0 = 0; k0 < K; k0 += 16) {
        __syncthreads();   // protect LDS from previous iteration's readers

        // --- stage A tile: 64 rows x 16 cols, 256 float4 loads / 128 threads
#pragma unroll
        for (int i = 0; i < 2; ++i) {
            const int idx = t + i * 128;
            const int r   = idx >> 2;
            const int c   = (idx & 3) << 2;
            const int gr  = bm + r;
            float4 v = make_float4(0.f, 0.f, 0.f, 0.f);
            if (gr < M)
                v = *reinterpret_cast<const float4*>(A + (size_t)gr * K + k0 + c);
            *reinterpret_cast<float4*>(&As[r * 20 + c]) = v;
        }
        // --- stage B tile: 16 rows x 64 cols
#pragma unroll
        for (int i = 0; i < 2; ++i) {
            const int idx = t + i * 128;
            const int kr  = idx >> 4;
            const int c   = (idx & 15) << 2;
            float4 v = *reinterpret_cast<const float4*>(B + (size_t)(k0 + kr) * N + bn + c);
            *reinterpret_cast<float4*>(&Bs[kr * 72 + c]) = v;
        }
        __syncthreads();

        // --- 4 WMMA k-steps of 4 over the 16-wide K chunk
#pragma unroll
        for (int kk = 0; kk < 16; kk += 4) {
            const int ka = kk + 2 * half;          // this half-wave's K columns
            v2f a;
            a.x = As[aRowBase * 20 + ka];
            a.y = As[aRowBase * 20 + ka + 1];
#pragma unroll
            for (int s = 0; s < 4; ++s) {
                v2f b;
                b.x = Bs[ka * 72 + s * 16 + lm];
                b.y = Bs[(ka + 1) * 72 + s * 16 + lm];
                acc[s] = __builtin_amdgcn_wmma_f32_16x16x4_f32(
                    false, a, false, b, (short)0, acc[s], false, false);
            }
        }
    }

    // --- store: acc VGPR r -> row (wave*16 + r + 8*half), col bn + s*16 + lm
#pragma unroll
    for (int r = 0; r < 8; ++r) {
        const int gm = bm + wave * 16 + r + 8 * half;
        if (gm < M) {
            float* cp = C + (size_t)gm * N + bn + lm;
#pragma unroll
            for (int s = 0; s < 4; ++s) cp[s * 16] = acc[s][r];
        }
    }
}

// ---------------------------------------------------------------------------
// SpMM scatter: agg[row] += val * h[col], 512 features per edge.
// One edge per 128-thread block; each thread handles a float4 of features.
// Uses non-returning global_atomic_add_f32 (bandwidth-bound path).
// ---------------------------------------------------------------------------
__global__ __launch_bounds__(128)
void spmm512_scatter(const int* __restrict__ rows, const int* __restrict__ cols,
                     const float* __restrict__ vals, const float* __restrict__ h,
                     float* __restrict__ agg, int nE)
{
    const int e = blockIdx.x;
    if (e >= nE) return;
    const int   row = rows[e];
    const int   col = cols[e];
    const float val = vals[e];
    const int f = threadIdx.x << 2;
    const float4 hv = *reinterpret_cast<const float4*>(h + (size_t)col * 512 + f);
    float* out = agg + (size_t)row * 512 + f;
    atomicAdd(out + 0, val * hv.x);
    atomicAdd(out + 1, val * hv.y);
    atomicAdd(out + 2, val * hv.z);
    atomicAdd(out + 3, val * hv.w);
}

// ---------------------------------------------------------------------------
// BatchNorm pass 1: per-column sum / sumsq partial reduction into stats
// (stats[0..C) = sum, stats[C..2C) = sumsq; must be pre-zeroed).
// ---------------------------------------------------------------------------
__global__ __launch_bounds__(256)
void bn_stats(const float* __restrict__ H, float* __restrict__ stats, int M, int C)
{
    const int t = threadIdx.x;
    const int rowsPerBlk = (M + gridDim.x - 1) / gridDim.x;
    const int r0 = blockIdx.x * rowsPerBlk;
    const int r1 = min(M, r0 + rowsPerBlk);

    float s0 = 0.f, q0 = 0.f, s1 = 0.f, q1 = 0.f;
    for (int r = r0; r < r1; ++r) {
        const float* p = H + (size_t)r * C;
        const float x = p[t];
        s0 += x; q0 += x * x;
        if (C > 256) {
            const float y = p[t + 256];
            s1 += y; q1 += y * y;
        }
    }
    atomicAdd(&stats[t], s0);
    atomicAdd(&stats[C + t], q0);
    if (C > 256) {
        atomicAdd(&stats[t + 256], s1);
        atomicAdd(&stats[C + t + 256], q1);
    }
}

// BatchNorm pass 2: fold mean/var/gamma/beta into per-column scale & shift.
__global__ void bn_finalize(const float* __restrict__ stats,
                            const float* __restrict__ gamma, const float* __restrict__ beta,
                            float* __restrict__ scale, float* __restrict__ shift,
                            int M, int C)
{
    const int c = blockIdx.x * blockDim.x + threadIdx.x;
    if (c < C) {
        const float invM = 1.0f / (float)M;
        const float mean = stats[c] * invM;
        const float var  = stats[C + c] * invM - mean * mean;
        const float sc   = gamma[c] * rsqrtf(var + 1e-5f);
        scale[c] = sc;
        shift[c] = beta[c] - mean * sc;
    }
}

// BatchNorm apply + numerically-stable softplus (in-place safe).
__global__ __launch_bounds__(256)
void bn_apply_softplus(const float* __restrict__ X, float* __restrict__ Y,
                       const float* __restrict__ scale, const float* __restrict__ shift,
                       size_t total, int cmask)
{
    size_t i = (size_t)blockIdx.x * blockDim.x + threadIdx.x;
    const size_t stride = (size_t)gridDim.x * blockDim.x;
    for (; i < total; i += stride) {
        const int c = (int)i & cmask;
        const float x  = X[i] * scale[c] + shift[c];
        const float sp = fmaxf(x, 0.f) + log1pf(__expf(-fabsf(x)));
        Y[i] = sp;
    }
}

// ---------------------------------------------------------------------------
extern "C" void kernel_launch(void* const* d_in, const int* in_sizes, int n_in,
                              void* d_out, int out_size, void* d_ws, size_t ws_size,
                              hipStream_t stream)
{
    const float* x      = (const float*)d_in[0];
    const int*   arows  = (const int*)  d_in[1];
    const int*   acols  = (const int*)  d_in[2];
    const float* avals  = (const float*)d_in[3];
    const float* Wg     = (const float*)d_in[4];
    const float* bg     = (const float*)d_in[5];
    const float* W1     = (const float*)d_in[6];
    const float* b1     = (const float*)d_in[7];
    const float* W21    = (const float*)d_in[8];
    const float* b21    = (const float*)d_in[9];
    const float* W22    = (const float*)d_in[10];
    const float* b22    = (const float*)d_in[11];
    const float* gamma0 = (const float*)d_in[12];
    const float* beta0  = (const float*)d_in[13];
    const float* gamma1 = (const float*)d_in[14];
    const float* beta1  = (const float*)d_in[15];

    const int N  = 50000, IN_DIM = 2000, GD = 512, HD = 256, LD = 64;
    const int NE = in_sizes[1];

    // Workspace layout (floats). h buffer is reused for hmid after SpMM.
    float* ws   = (float*)d_ws;
    float* h    = ws;                         // N*GD  (GEMM1 out; later hmid N*HD)
    float* agg  = ws + (size_t)N * GD;        // N*GD  (SpMM out; h0 in-place)
    float* st   = agg + (size_t)N * GD;       // small stats block (3072 floats)
    float* sum0 = st;            float* sc0 = st + 1024; float* sh0 = st + 1536;
    float* sum1 = st + 2048;     float* sc1 = st + 2560; float* sh1 = st + 2816;

    // Deterministic per-call init (harness does not re-poison between replays).
    hipMemsetAsync(agg, 0, (size_t)N * GD * sizeof(float), stream);
    hipMemsetAsync(st,  0, 3072 * sizeof(float), stream);

    const dim3 blk(128);
    const int mblks = (N + 63) / 64;

    // GCN dense projection: h = x @ Wg + bg
    gemm_bias_wmma_f32<<<dim3(mblks, GD / 64), blk, 0, stream>>>(x, Wg, bg, h, N, GD, IN_DIM);

    // Sparse aggregation
    spmm512_scatter<<<NE, 128, 0, stream>>>(arows, acols, avals, h, agg, NE);

    // BN0 + softplus (h0 in-place in agg)
    bn_stats<<<512, 256, 0, stream>>>(agg, sum0, N, GD);
    bn_finalize<<<2, 256, 0, stream>>>(sum0, gamma0, beta0, sc0, sh0, N, GD);
    bn_apply_softplus<<<4096, 256, 0, stream>>>(agg, agg, sc0, sh0, (size_t)N * GD, GD - 1);

    // hmid = h0 @ W1 + b1 (reuse h buffer)
    float* hmid = h;
    gemm_bias_wmma_f32<<<dim3(mblks, HD / 64), blk, 0, stream>>>(agg, W1, b1, hmid, N, HD, GD);

    // BN1 + softplus (h1 in-place in hmid)
    bn_stats<<<512, 256, 0, stream>>>(hmid, sum1, N, HD);
    bn_finalize<<<1, 256, 0, stream>>>(sum1, gamma1, beta1, sc1, sh1, N, HD);
    bn_apply_softplus<<<4096, 256, 0, stream>>>(hmid, hmid, sc1, sh1, (size_t)N * HD, HD - 1);

    // mu / logvar heads -> d_out (concatenated)
    float* mu = (float*)d_out;
    float* lv = mu + (size_t)N * LD;
    gemm_bias_wmma_f32<<<dim3(mblks, LD / 64), blk, 0, stream>>>(hmid, W21, b21, mu, N, LD, HD);
    gemm_bias_wmma_f32<<<dim3(mblks, LD / 64), blk, 0, stream>>>(hmid, W22, b22, lv, N, LD, HD);
}